// RoutingLayer_59382217834805
// MI455X (gfx1250) — compile-verified
//
#include <hip/hip_runtime.h>
#include <math.h>

// Problem constants (reference: B=64, K=16, J=512, I=128)
#define B_ 64
#define K_ 16
#define J_ 512
#define I_ 128
#define SCALE_F 0.08838834764831845f   // 1/sqrt(128)
#define EPS_F   1e-20f

typedef __attribute__((ext_vector_type(2))) float v2f;
typedef __attribute__((ext_vector_type(8))) float v8f;

// ---------------------------------------------------------------------------
// Kernel 1: logits[k][j][b] = SCALE * sum_i (x[b,j,i]*w[k,j,i])^2
// Per j this is a 64x16 = Xsq(64x128) @ Wsq^T(128x16) GEMM -> V_WMMA_F32_16X16X4_F32.
// One wave (32 lanes) per j; 4 batch tiles of 16 rows; 32 K-steps of 4.
// f32 WMMA keeps the softmax logits exact; at 0.13 GFLOP total this is free.
// ---------------------------------------------------------------------------
__global__ __launch_bounds__(32) void logits_wmma(const float* __restrict__ x,
                                                  const float* __restrict__ w,
                                                  float* __restrict__ logits) {
    const int j    = blockIdx.x;
    const int lane = threadIdx.x;      // wave32: EXEC all 1s, no divergence
    const int m    = lane & 15;        // A: batch row within tile / B: capsule N
    const int hi   = lane >> 4;        // K-half selector: K pair = {2*hi, 2*hi+1}

    // A rows: batch b = mt*16 + m ; B rows: capsule n = m
    const float* __restrict__ xb = x + (size_t)m * (J_ * I_) + (size_t)j * I_;
    const float* __restrict__ wb = w + (size_t)m * (J_ * I_) + (size_t)j * I_;
    const size_t tstride = (size_t)16 * J_ * I_;   // 16 batch rows per M-tile

    v8f acc[4];
    acc[0] = (v8f){}; acc[1] = (v8f){}; acc[2] = (v8f){}; acc[3] = (v8f){};

    for (int ki = 0; ki < I_; ki += 4) {
        const int k0 = ki + 2 * hi;
        // B tile: B[k][n] = w[n,j,k]^2  (lane%16 = n, K-half by lane group)
        float w0 = wb[k0], w1 = wb[k0 + 1];
        v2f bm; bm.x = w0 * w0; bm.y = w1 * w1;
#pragma unroll
        for (int mt = 0; mt < 4; ++mt) {
            float x0 = xb[(size_t)mt * tstride + k0];
            float x1 = xb[(size_t)mt * tstride + k0 + 1];
            v2f am; am.x = x0 * x0; am.y = x1 * x1;
            acc[mt] = __builtin_amdgcn_wmma_f32_16x16x4_f32(
                /*neg_a=*/false, am, /*neg_b=*/false, bm,
                /*c_mod=*/(short)0, acc[mt],
                /*reuse_a=*/false, /*reuse_b=*/false);
        }
    }

    // C/D layout: VGPR v, lanes 0-15 -> M=v, lanes 16-31 -> M=v+8; N = lane%16.
    // Store as (K, J, B) so the batch-softmax reads contiguously.
#pragma unroll
    for (int mt = 0; mt < 4; ++mt) {
#pragma unroll
        for (int v = 0; v < 8; ++v) {
            const int brow = mt * 16 + v + 8 * hi;
            logits[((size_t)m * J_ + j) * B_ + brow] = acc[mt][v] * SCALE_F;
        }
    }
}

// ---------------------------------------------------------------------------
// Kernel 2: softmax over the batch dim (axis=0) for each (k,j), plus bias.
// One block of 64 threads per (k,j) column; max-subtracted like jax.nn.softmax.
// ---------------------------------------------------------------------------
__global__ __launch_bounds__(64) void softmax_batch(float* __restrict__ c,
                                                    const float* __restrict__ bias) {
    const int kj = blockIdx.x;          // k*J + j
    const int t  = threadIdx.x;
    __shared__ float sh[B_];

    const float v = c[(size_t)kj * B_ + t];
    sh[t] = v;
    __syncthreads();
#pragma unroll
    for (int off = 32; off > 0; off >>= 1) {
        if (t < off) sh[t] = fmaxf(sh[t], sh[t + off]);
        __syncthreads();
    }
    const float mx = sh[0];
    __syncthreads();

    const float e = __expf(v - mx);
    sh[t] = e;
    __syncthreads();
#pragma unroll
    for (int off = 32; off > 0; off >>= 1) {
        if (t < off) sh[t] += sh[t + off];
        __syncthreads();
    }
    const float s = sh[0];

    c[(size_t)kj * B_ + t] = e / s + bias[kj];
}

// ---------------------------------------------------------------------------
// Kernel 3: s[b,k,i] = sum_j c[b,k,j] * x[b,j,i] * w[k,j,i], fused squash.
// Block = (b,k), thread = i. x/W stream from L2 (whole problem is L2-resident).
// LDS tree-reduce for ||s||, then out = (1 - 1/(exp(n)+eps)) * s/(n+eps).
// Writes d_out fully every call (deterministic under graph replay).
// ---------------------------------------------------------------------------
__global__ __launch_bounds__(128) void sumsq_squash(const float* __restrict__ x,
                                                    const float* __restrict__ w,
                                                    const float* __restrict__ c,
                                                    float* __restrict__ out) {
    const int bk = blockIdx.x;          // b*K + k
    const int b  = bk / K_;
    const int k  = bk % K_;
    const int i  = threadIdx.x;

    const float* __restrict__ xp = x + (size_t)b * J_ * I_ + i;
    const float* __restrict__ wp = w + (size_t)k * J_ * I_ + i;
    const float* __restrict__ cp = c + (size_t)k * J_ * B_ + b;   // stride B_ per j

    float acc = 0.0f;
#pragma unroll 4
    for (int j = 0; j < J_; ++j) {
        const float cj = cp[(size_t)j * B_];          // block-uniform -> scalar load
        acc = fmaf(cj * xp[(size_t)j * I_], wp[(size_t)j * I_], acc);
    }

    __shared__ float red[I_];
    red[i] = acc * acc;
    __syncthreads();
#pragma unroll
    for (int off = I_ / 2; off > 0; off >>= 1) {
        if (i < off) red[i] += red[i + off];
        __syncthreads();
    }
    const float n = sqrtf(red[0]);
    const float factor = (1.0f - 1.0f / (__expf(n) + EPS_F)) / (n + EPS_F);

    out[(size_t)bk * I_ + i] = factor * acc;
}

// ---------------------------------------------------------------------------
// d_in[0]=inputs (B,J,I) f32, d_in[1]=W (K,J,I) f32, d_in[2]=b (K,J,1) f32.
// d_out = (B,K,I) f32. Scratch: logits/c buffer K*J*B floats = 2 MB in d_ws.
// ---------------------------------------------------------------------------
extern "C" void kernel_launch(void* const* d_in, const int* in_sizes, int n_in,
                              void* d_out, int out_size, void* d_ws, size_t ws_size,
                              hipStream_t stream) {
    const float* x    = (const float*)d_in[0];
    const float* w    = (const float*)d_in[1];
    const float* bias = (const float*)d_in[2];
    float* out    = (float*)d_out;
    float* logits = (float*)d_ws;      // needs K_*J_*B_*4 = 2 MB of scratch

    logits_wmma <<<J_,      32, 0, stream>>>(x, w, logits);
    softmax_batch<<<K_ * J_, 64, 0, stream>>>(logits, bias);
    sumsq_squash <<<B_ * K_, 128, 0, stream>>>(x, w, logits, out);
}